// PredictionHead_6244882448979
// MI455X (gfx1250) — compile-verified
//
#include <hip/hip_runtime.h>

typedef __attribute__((ext_vector_type(2))) float v2f;
typedef __attribute__((ext_vector_type(8))) float v8f;

#define SEQ      4096
#define BATCH    16
#define UPPER    30                  // band: 0 <= j - i <= 30 (window width 31)
#define THREADS  512
#define WAVES    (THREADS / 32)      // 16 wave32 per block
#define PER_THR  (SEQ / THREADS)     // 8 elements per thread
#define OFS      30                  // leading zero pad for shifted start row
#define PADDED   (SEQ + 64)          // room for tile overshoot (base+39 max = 4127)

// ---------------------------------------------------------------------------
// Kernel A: row softmax. One block per (tensor, row): grid = 32 blocks.
// b128 loads, wave32 shfl_xor max, exp in registers, denominator via chained
// V_WMMA_F32_16X16X4_F32 against an all-ones B (exact fp32 wave reduction).
// ---------------------------------------------------------------------------
__global__ __launch_bounds__(THREADS) void softmax_rows(
    const float* __restrict__ start_logits,
    const float* __restrict__ end_logits,
    float* __restrict__ out)
{
  const int row    = blockIdx.x & (BATCH - 1);
  const int tensor = blockIdx.x >> 4;            // 0 = start, 1 = end
  const float* src = (tensor ? end_logits : start_logits) + row * SEQ;
  float* dst       = out + tensor * (BATCH * SEQ) + row * SEQ;

  const int tid  = threadIdx.x;
  const int lane = tid & 31;
  const int wid  = tid >> 5;

  __shared__ float red[WAVES];

  const float4* src4 = (const float4*)(src + tid * PER_THR);
  float4 x0 = src4[0];
  float4 x1 = src4[1];

  // ---- row max: thread-local -> wave shfl_xor -> cross-wave via LDS ----
  float m = fmaxf(fmaxf(fmaxf(x0.x, x0.y), fmaxf(x0.z, x0.w)),
                  fmaxf(fmaxf(x1.x, x1.y), fmaxf(x1.z, x1.w)));
#pragma unroll
  for (int d = 16; d > 0; d >>= 1) m = fmaxf(m, __shfl_xor(m, d, 32));
  if (lane == 0) red[wid] = m;
  __syncthreads();
  float rowmax = red[0];
#pragma unroll
  for (int w = 1; w < WAVES; ++w) rowmax = fmaxf(rowmax, red[w]);
  __syncthreads();   // red[] reused below

  // ---- exp(x - max), kept in registers ----
  v8f e;
  e[0] = __expf(x0.x - rowmax); e[1] = __expf(x0.y - rowmax);
  e[2] = __expf(x0.z - rowmax); e[3] = __expf(x0.w - rowmax);
  e[4] = __expf(x1.x - rowmax); e[5] = __expf(x1.y - rowmax);
  e[6] = __expf(x1.z - rowmax); e[7] = __expf(x1.w - rowmax);

  // ---- wave-exact fp32 sum via V_WMMA_F32_16X16X4_F32, B = ones:
  //      D[m,n] = sum_k A[m,k] + C[m,n]  (A covers 2 values/lane per WMMA) ----
  v8f acc = {};
  const v2f ones = {1.0f, 1.0f};
#pragma unroll
  for (int k = 0; k < 4; ++k) {
    v2f a;
    a[0] = e[2 * k];
    a[1] = e[2 * k + 1];
    acc = __builtin_amdgcn_wmma_f32_16x16x4_f32(
        /*neg_a=*/false, a, /*neg_b=*/false, ones,
        /*c_mod=*/(short)0, acc, /*reuse_a=*/false, /*reuse_b=*/false);
  }
  float ws = ((acc[0] + acc[1]) + (acc[2] + acc[3])) +
             ((acc[4] + acc[5]) + (acc[6] + acc[7]));
  ws += __shfl_xor(ws, 16, 32);   // full wave sum in every lane

  if (lane == 0) red[wid] = ws;
  __syncthreads();
  float total = 0.0f;
#pragma unroll
  for (int w = 0; w < WAVES; ++w) total += red[w];
  const float inv = 1.0f / total;

  float4* dst4 = (float4*)(dst + tid * PER_THR);
  dst4[0] = make_float4(e[0] * inv, e[1] * inv, e[2] * inv, e[3] * inv);
  dst4[1] = make_float4(e[4] * inv, e[5] * inv, e[6] * inv, e[7] * inv);
}

// ---------------------------------------------------------------------------
// 8 sliding-window (width 31) maxes over a 40-element register tile:
//   w[k] = max(t[k .. k+30]) = max( suffix(t[k..7]), prefix(t[8..k+30]) )
// 44 v_max ops instead of 8*31 = 248.
// ---------------------------------------------------------------------------
__device__ __forceinline__ void windows31(const float (&t)[40], float (&w)[8]) {
  float S[8];
  S[7] = t[7];
#pragma unroll
  for (int k = 6; k >= 0; --k) S[k] = fmaxf(t[k], S[k + 1]);
  float p = t[8];
#pragma unroll
  for (int m = 9; m <= 30; ++m) p = fmaxf(p, t[m]);
  w[0] = fmaxf(S[0], p);
#pragma unroll
  for (int k = 1; k < 8; ++k) {
    p = fmaxf(p, t[30 + k]);
    w[k] = fmaxf(S[k], p);
  }
}

__device__ __forceinline__ void load_tile(const float* arr, int base, float (&t)[40]) {
  const float4* p4 = (const float4*)(arr + base);
#pragma unroll
  for (int q = 0; q < 10; ++q) {
    float4 v = p4[q];
    t[4 * q + 0] = v.x; t[4 * q + 1] = v.y;
    t[4 * q + 2] = v.z; t[4 * q + 3] = v.w;
  }
}

// ---------------------------------------------------------------------------
// Kernel B: banded argmax. One block per batch row; both prob rows LDS-resident
// with zero padding (exactly matches the reference's masked-to-0.0 semantics,
// so all bounds checks vanish). start row stored shifted by OFS=30 so the
// backward window sp[i-30..i] becomes the forward window sp_pad[i..i+30].
// ---------------------------------------------------------------------------
__global__ __launch_bounds__(THREADS) void band_argmax(
    const float* __restrict__ probs,   // [2*BATCH*SEQ]: start then end
    float* __restrict__ out_ptr)       // [2*BATCH]: start ptrs then end ptrs
{
  const int b = blockIdx.x;
  const float* sp_g = probs + b * SEQ;
  const float* ep_g = probs + BATCH * SEQ + b * SEQ;

  __shared__ __align__(16) float ep_pad[PADDED];   // ep_pad[i] = end_prob[i], 0 beyond
  __shared__ __align__(16) float sp_pad[PADDED];   // sp_pad[OFS+i] = start_prob[i], 0 pads
  __shared__ float rsv[WAVES]; __shared__ int rsi[WAVES];
  __shared__ float rev[WAVES]; __shared__ int rei[WAVES];

  const int tid  = threadIdx.x;
  const int lane = tid & 31;
  const int wid  = tid >> 5;
  const int base = tid * PER_THR;

  // ---- fill LDS (ep aligned b128; shifted sp is 8B-aligned -> b64) ----
  {
    const float4* sg = (const float4*)(sp_g + base);
    const float4* eg = (const float4*)(ep_g + base);
    float4 s0 = sg[0], s1 = sg[1];
    float4 e0 = eg[0], e1 = eg[1];
    float4* ed = (float4*)(ep_pad + base);
    ed[0] = e0; ed[1] = e1;
    float2* sd = (float2*)(sp_pad + OFS + base);
    sd[0] = make_float2(s0.x, s0.y);
    sd[1] = make_float2(s0.z, s0.w);
    sd[2] = make_float2(s1.x, s1.y);
    sd[3] = make_float2(s1.z, s1.w);
  }
  if (tid < OFS) sp_pad[tid] = 0.0f;                                 // leading pad
  if (tid < PADDED - SEQ) ep_pad[SEQ + tid] = 0.0f;                  // 64 trailing
  if (tid < PADDED - (OFS + SEQ)) sp_pad[OFS + SEQ + tid] = 0.0f;    // 34 trailing
  __syncthreads();

  float t[40];
  float w[8];

  // ---- start side: score(i) = start_prob[i] * max(end_prob[i .. i+30]) ----
  load_tile(ep_pad, base, t);
  windows31(t, w);
  float spv[8];
  {
    const float2* sv2 = (const float2*)(sp_pad + OFS + base);
#pragma unroll
    for (int q = 0; q < 4; ++q) {
      float2 v = sv2[q];
      spv[2 * q] = v.x; spv[2 * q + 1] = v.y;
    }
  }
  float bsv = -1.0f; int bsi = 0;
#pragma unroll
  for (int k = 0; k < PER_THR; ++k) {            // ascending k keeps first index on ties
    const float sc = spv[k] * w[k];
    if (sc > bsv) { bsv = sc; bsi = base + k; }
  }

  // ---- end side: score(j) = end_prob[j] * max(start_prob[j-30 .. j])
  //      = end_prob[j] * max(sp_pad[j .. j+30]) ----
  load_tile(sp_pad, base, t);
  windows31(t, w);
  float epv[8];
  {
    const float4* ev4 = (const float4*)(ep_pad + base);
    float4 v0 = ev4[0], v1 = ev4[1];
    epv[0] = v0.x; epv[1] = v0.y; epv[2] = v0.z; epv[3] = v0.w;
    epv[4] = v1.x; epv[5] = v1.y; epv[6] = v1.z; epv[7] = v1.w;
  }
  float bev = -1.0f; int bei = 0;
#pragma unroll
  for (int k = 0; k < PER_THR; ++k) {
    const float sc = epv[k] * w[k];
    if (sc > bev) { bev = sc; bei = base + k; }
  }

  // ---- wave32 (value,index) argmax reduction, first-index tie-break ----
#pragma unroll
  for (int d = 16; d > 0; d >>= 1) {
    float ov = __shfl_xor(bsv, d, 32); int oi = __shfl_xor(bsi, d, 32);
    if (ov > bsv || (ov == bsv && oi < bsi)) { bsv = ov; bsi = oi; }
    float ow = __shfl_xor(bev, d, 32); int oj = __shfl_xor(bei, d, 32);
    if (ow > bev || (ow == bev && oj < bei)) { bev = ow; bei = oj; }
  }
  if (lane == 0) { rsv[wid] = bsv; rsi[wid] = bsi; rev[wid] = bev; rei[wid] = bei; }
  __syncthreads();

  if (tid == 0) {
    float v = rsv[0]; int ix = rsi[0];
#pragma unroll
    for (int w2 = 1; w2 < WAVES; ++w2)
      if (rsv[w2] > v || (rsv[w2] == v && rsi[w2] < ix)) { v = rsv[w2]; ix = rsi[w2]; }
    out_ptr[b] = (float)ix;

    float v2 = rev[0]; int ix2 = rei[0];
#pragma unroll
    for (int w2 = 1; w2 < WAVES; ++w2)
      if (rev[w2] > v2 || (rev[w2] == v2 && rei[w2] < ix2)) { v2 = rev[w2]; ix2 = rei[w2]; }
    out_ptr[BATCH + b] = (float)ix2;
  }
}

extern "C" void kernel_launch(void* const* d_in, const int* in_sizes, int n_in,
                              void* d_out, int out_size, void* d_ws, size_t ws_size,
                              hipStream_t stream) {
  const float* start_logits = (const float*)d_in[0];
  const float* end_logits   = (const float*)d_in[1];
  float* out = (float*)d_out;

  // out layout (floats): [0,65536) start_prob | [65536,131072) end_prob |
  //                      [131072,131088) start_pointer | [131088,131104) end_pointer
  softmax_rows<<<dim3(2 * BATCH), dim3(THREADS), 0, stream>>>(start_logits, end_logits, out);
  band_argmax<<<dim3(BATCH), dim3(THREADS), 0, stream>>>(out, out + 2 * BATCH * SEQ);
}